// DifferentiableOscillatorBank_13795434955033
// MI455X (gfx1250) — compile-verified
//
#include <hip/hip_runtime.h>
#include <math.h>

#define SAMPLES   88200
#define BATCH     16
#define OSC       32
#define CST_STRIDE 16                 // floats per oscillator record (16B-aligned vectors)
#define TWO_PI_F  6.28318530717958647692f
#define SR_F      44100.0f

typedef float v2f __attribute__((ext_vector_type(2)));
typedef float v8f __attribute__((ext_vector_type(8)));

#if defined(__has_builtin)
#  if __has_builtin(__builtin_amdgcn_global_load_async_to_lds_b32) && \
      __has_builtin(__builtin_amdgcn_s_wait_asynccnt)
#    define OSC_ASYNC_LDS 1
#  endif
#  if __has_builtin(__builtin_amdgcn_exp2f)
#    define OSC_EXP2(x) __builtin_amdgcn_exp2f(x)
#  else
#    define OSC_EXP2(x) exp2f(x)
#  endif
#  if __has_builtin(__builtin_amdgcn_sinf)
     // v_sin_f32 input unit is revolutions (disasm-confirmed: OCML pre-multiplies
     // by 1/2pi); we already have the fractional revolution, so feed it directly.
#    define OSC_SIN_REV(fr) __builtin_amdgcn_sinf(fr)
#  else
#    define OSC_SIN_REV(fr) __sinf(TWO_PI_F * (fr))
#  endif
#else
#  define OSC_EXP2(x)      exp2f(x)
#  define OSC_SIN_REV(fr)  __sinf(TWO_PI_F * (fr))
#endif

#ifdef OSC_ASYNC_LDS
typedef __attribute__((address_space(1))) int async_gint;   // global (device) int
typedef __attribute__((address_space(3))) int async_lint;   // LDS int
#endif

// ---------------------------------------------------------------------------
// Pass 1: per-(batch, oscillator) closed-form constants.
//   record layout (floats):
//   0 max_amp | 1 1/L | 2 L | 3 log2(adr) | 4 F/SR | 5 s_start | 6 phase(rev) |
//   7 ln(r)   | 8 1/(1-r) | 9 r^(s_start-L) | 10..15 pad
// r==1 limit encoded as lnr=-2^-60, inv_omr=2^60 so that
//   -expm1(n*lnr)*inv_omr == n exactly (power-of-2 scaling) -> branchless.
// ---------------------------------------------------------------------------
__global__ __launch_bounds__(512)
void osc_precompute(const float* __restrict__ params,
                    const float* __restrict__ logits,
                    const float* __restrict__ notef,
                    const float* __restrict__ vel,
                    float* __restrict__ cst)
{
    const int i = blockIdx.x * blockDim.x + threadIdx.x;
    if (i >= BATCH * OSC) return;
    const int b = i >> 5;
    const float* p = params + i * 7;

    const float amp_factor = p[0];
    const float ffn        = p[1];
    const float phasef     = p[2];
    const float adf        = p[3];
    const float aaf        = p[4];
    const float fdf        = p[5];
    const float cplp       = p[6];

    const float act = 1.0f / (1.0f + expf(-logits[i]));

    // amplitude decay rate
    const float K_MAX_AD = 0.99921442756f;
    const float K_MIN_AD = 0.99999842823f;
    const float adr = K_MAX_AD + (K_MIN_AD - K_MAX_AD) * adf;

    // frequency decay: r = 1 - omr, omr computed WITHOUT cancellation
    const float OMR_MAX = (float)(1.0 - 0.99999991268);   // 8.732e-08
    const float omr = fmaxf(OMR_MAX * (1.0f - fdf), 0.0f);
    float lnr, inv_omr;
    if (omr > 0.0f) {
        lnr     = log1pf(-omr);       // ln(r), stable near r = 1
        inv_omr = 1.0f / omr;
    } else {                          // r == 1 limit: G(n) = n exactly
        lnr     = -0x1p-60f;
        inv_omr =  0x1p+60f;
    }

    // attack length
    const float K_SI      = 1.0f / SR_F;
    const float K_MAX_AAR = 20.0f * K_SI;
    const float K_MIN_AAR = K_SI / 150.0f;
    const float aar = K_MIN_AAR + (K_MAX_AAR - K_MIN_AAR) * aaf;
    const float L   = fmaxf(1.0f / fmaxf(aar, 1e-8f), 1.0f);

    // frequency factor decode (anchors are 2^(ai-1): 0.5..32)
    float x  = fminf(fmaxf(ffn, 0.0f), 1.0f) * 7.0f;
    int   ai = (int)floorf(x);
    ai = ai < 0 ? 0 : (ai > 6 ? 6 : ai);
    const float local  = fminf(fmaxf(x - (float)ai, 0.0f), 1.0f);
    const float cp     = fminf(fmaxf(cplp, 0.0f), 1.0f);
    const float base   = exp2f((float)(ai - 1));
    const float detune = 1.0f + (local - 0.5f) * 2.0f * (cp * 0.025f + (1.0f - cp) * 0.05f);
    const float FMAX   = SR_F * 0.5f - 1.0f;
    const float F      = fminf(fmaxf(notef[b] * base * detune, 0.0f), FMAX);

    const float s_start = floorf(L) + 1.0f;      // first integer t with t > L
    const float d       = s_start - L;           // fractional exponent in (0,1]
    const float rd      = expf(d * lnr);

    float* c = cst + i * CST_STRIDE;
    c[0]  = vel[b] * amp_factor * act;
    c[1]  = 1.0f / L;
    c[2]  = L;
    c[3]  = log2f(adr);
    c[4]  = F * (1.0f / SR_F);        // frequency in revolutions/sample
    c[5]  = s_start;
    c[6]  = phasef;                   // phase offset in revolutions
    c[7]  = lnr;
    c[8]  = inv_omr;
    c[9]  = rd;
    c[10] = 0.0f; c[11] = 0.0f; c[12] = 0.0f; c[13] = 0.0f; c[14] = 0.0f; c[15] = 0.0f;
}

// ---------------------------------------------------------------------------
// Pass 2: render.  8 waves/block; each wave renders a 16-sample tile.
// Lane n (=lane&15) owns time t0+n; the 32-oscillator mix is accumulated in
// the matrix pipe: D = ones(16x4) x B(4x16) + C, chained over 8 WMMAs.
// ---------------------------------------------------------------------------
__global__ __launch_bounds__(256)
void osc_render(const float* __restrict__ cst, float* __restrict__ out)
{
    __shared__ __align__(16) float sc[OSC * CST_STRIDE];   // 2 KB constant table
    const int  b    = blockIdx.y;
    const float* g  = cst + (size_t)b * OSC * CST_STRIDE;
    const int  tid  = threadIdx.x;

#ifdef OSC_ASYNC_LDS
    __builtin_amdgcn_global_load_async_to_lds_b32(
        (async_gint*)(g + tid), (async_lint*)(sc + tid), 0, 0);
    __builtin_amdgcn_global_load_async_to_lds_b32(
        (async_gint*)(g + tid + 256), (async_lint*)(sc + tid + 256), 0, 0);
    __builtin_amdgcn_s_wait_asynccnt(0);
#else
    sc[tid]       = g[tid];
    sc[tid + 256] = g[tid + 256];
#endif
    __syncthreads();

    const int wave = tid >> 5;
    const int lane = tid & 31;
    const int n    = lane & 15;
    const int h    = lane >> 4;

    const int t0 = (blockIdx.x * 8 + wave) * 16;
    if (t0 >= SAMPLES) return;                 // uniform per wave: EXEC stays full

    const int   ti    = t0 + n;
    const float t     = (float)ti;
    const float tp1   = t + 1.0f;
    const bool  valid = (ti < SAMPLES);

    v8f acc  = {0.f, 0.f, 0.f, 0.f, 0.f, 0.f, 0.f, 0.f};
    v2f ones = {1.0f, 1.0f};

    #pragma unroll
    for (int i = 0; i < 8; ++i) {
        v2f bmat = {0.f, 0.f};
        #pragma unroll
        for (int j = 0; j < 2; ++j) {
            const int o = (i << 2) + (h << 1) + j;   // each of 32 oscillators exactly once
            const float4 q0 = *(const float4*)&sc[o * CST_STRIDE + 0]; // max_amp,1/L,L,log2(adr)
            const float4 q1 = *(const float4*)&sc[o * CST_STRIDE + 4]; // F/SR,s_start,ph,lnr
            const float2 q2 = *(const float2*)&sc[o * CST_STRIDE + 8]; // 1/(1-r), r^d

            // amplitude envelope (native v_exp_f32; exponent in [-101, 0])
            float prog  = fminf(tp1 * q0.y, 1.0f);
            float dstep = fmaxf(t - q0.z, 0.0f);
            float amp   = q0.x * prog * OSC_EXP2(dstep * q0.w);
            amp = valid ? amp : 0.0f;

            // closed-form phase integral; |x| <= 7.7e-3 in the live branch, so a
            // 3-FMA Taylor expm1 has rel. error < 2^-25 -> fully branchless
            float nn = t - q1.y + 1.0f;                  // samples past attack
            float xx = nn * q1.w;                        // n * ln(r)
            float pp = fmaf(xx, 0.166666667f, 0.5f);
            pp       = fmaf(xx, pp, 1.0f);
            float em = xx * pp;                          // expm1(n*ln r)
            float G  = -em * q2.x;                       // (1 - r^n)/(1 - r)
            float S  = (t < q1.y) ? tp1                  // still in attack
                                  : fmaf(q2.y, G, q1.y); // s_start + r^d * G
            // angle in revolutions -> frac in [0,1) -> native v_sin_f32 directly
            float rev = fmaf(q1.x, S, q1.z);
            float fr  = rev - floorf(rev);
            float v   = amp * OSC_SIN_REV(fr);
            if (j == 0) bmat.x = v; else bmat.y = v;
        }
        // D[m][n] = sum_k B[k][n] + C[m][n]  (A = all-ones)
        acc = __builtin_amdgcn_wmma_f32_16x16x4_f32(false, ones, false, bmat,
                                                    (short)0, acc, false, false);
    }

    // every row of D is the mix for time column n; row M=0 lives in VGPR0, lanes 0..15
    if (h == 0 && valid) {
        float m = fminf(fmaxf(acc[0], -1.0f), 1.0f);
        out[(size_t)b * SAMPLES + ti] = m;
    }
}

// ---------------------------------------------------------------------------
extern "C" void kernel_launch(void* const* d_in, const int* in_sizes, int n_in,
                              void* d_out, int out_size, void* d_ws, size_t ws_size,
                              hipStream_t stream)
{
    (void)in_sizes; (void)n_in; (void)out_size; (void)ws_size;

    const float* params = (const float*)d_in[0];   // (16,32,7)
    const float* logits = (const float*)d_in[1];   // (16,32)
    const float* notef  = (const float*)d_in[2];   // (16,)
    const float* vel    = (const float*)d_in[3];   // (16,)
    float* out = (float*)d_out;                    // (16,88200)
    float* cstw = (float*)d_ws;                    // needs 16*32*16*4 = 32 KB

    osc_precompute<<<dim3(1), dim3(512), 0, stream>>>(params, logits, notef, vel, cstw);

    const int ntiles = (SAMPLES + 15) / 16;        // 5513 16-sample tiles per batch
    const int gx     = (ntiles + 7) / 8;           // 8 tiles (waves) per block
    osc_render<<<dim3(gx, BATCH), dim3(256), 0, stream>>>(cstw, out);
}